// GINE5_20693152432431
// MI455X (gfx1250) — compile-verified
//
#include <hip/hip_runtime.h>
#include <hip/hip_bf16.h>

typedef __attribute__((ext_vector_type(2))) float v2f;
typedef __attribute__((ext_vector_type(8))) float v8f;

#define N_NODES  100000
#define N_EDGES  1600000
#define N_GRAPHS 2000
#define F_IN     32
#define EDGE_DIM 4
#define H        64
#define N_LAYERS 5

// ---------------------------------------------------------------------------
// Input projection: h = relu(x @ W_in + b_in), fp32 WMMA (16x16x4_f32).
// 4 waves/block, each wave computes a 16-row x 64-col tile.
// ---------------------------------------------------------------------------
__global__ __launch_bounds__(128) void gine_in_proj(
    const float* __restrict__ x, const float* __restrict__ W_in,
    const float* __restrict__ b_in, float* __restrict__ h) {
  __shared__ float WT[H * F_IN];  // WT[n*F_IN + k] = W_in[k*H + n] (transposed)
  __shared__ float bs[H];
  int tid = threadIdx.x;
  for (int i = tid; i < F_IN * H; i += 128) {
    int k = i / H, n = i % H;
    WT[n * F_IN + k] = W_in[i];
  }
  if (tid < H) bs[tid] = b_in[tid];
  __syncthreads();

  int wave = tid >> 5, lane = tid & 31;
  int m0 = blockIdx.x * 64 + wave * 16;
  if (m0 >= N_NODES) return;
  int mrow = lane & 15, half = lane >> 4;
  int row = m0 + mrow;

  v8f acc[4] = {};
  #pragma unroll
  for (int kk = 0; kk < F_IN / 4; ++kk) {
    int k = kk * 4 + half * 2;  // A frag: lanes 0-15 hold K={4kk,4kk+1}, 16-31 hold {4kk+2,4kk+3}
    v2f a = *(const v2f*)(x + row * F_IN + k);
    #pragma unroll
    for (int nt = 0; nt < 4; ++nt) {
      v2f b = *(const v2f*)(WT + (nt * 16 + mrow) * F_IN + k);
      acc[nt] = __builtin_amdgcn_wmma_f32_16x16x4_f32(
          false, a, false, b, (short)0, acc[nt], false, false);
    }
  }
  #pragma unroll
  for (int nt = 0; nt < 4; ++nt) {
    int n = nt * 16 + mrow;
    float bias = bs[n];
    #pragma unroll
    for (int r = 0; r < 8; ++r) {
      // C layout: VGPR r holds row (r + 8*half), col = lane%16 (+tile offset)
      h[(m0 + r + 8 * half) * H + n] = fmaxf(acc[nt][r] + bias, 0.0f);
    }
  }
}

// ---------------------------------------------------------------------------
// Edge phase (per layer): m = relu(h[src] + edge_attr @ We + be),
// agg[dst] += m.  4 lanes per edge, 16 features per lane. We/be in LDS.
// ---------------------------------------------------------------------------
__global__ __launch_bounds__(256) void gine_edge(
    const float* __restrict__ h, const int* __restrict__ src,
    const int* __restrict__ dst, const float* __restrict__ edge_attr,
    const float* __restrict__ We, const float* __restrict__ be,
    float* __restrict__ agg) {
  __shared__ float Ws[EDGE_DIM * H];
  __shared__ float bs[H];
  int tid = threadIdx.x;
  if (tid < EDGE_DIM * H) Ws[tid] = We[tid];
  if (tid < H) bs[tid] = be[tid];
  __syncthreads();

  int e = blockIdx.x * 64 + (tid >> 2);
  if (e >= N_EDGES) return;
  int q = tid & 3;  // feature quarter: j in [16q, 16q+16)
  int s = src[e], d = dst[e];
  float4 ea = *(const float4*)(edge_attr + e * 4);
  const float* hrow = h + s * H + q * 16;
  float* arow = agg + d * H + q * 16;

  #pragma unroll
  for (int jj = 0; jj < 4; ++jj) {
    int j = q * 16 + jj * 4;
    float4 hv = *(const float4*)(hrow + jj * 4);
    float m0 = fmaxf(hv.x + bs[j + 0] + ea.x * Ws[0 * H + j + 0] + ea.y * Ws[1 * H + j + 0]
                               + ea.z * Ws[2 * H + j + 0] + ea.w * Ws[3 * H + j + 0], 0.0f);
    float m1 = fmaxf(hv.y + bs[j + 1] + ea.x * Ws[0 * H + j + 1] + ea.y * Ws[1 * H + j + 1]
                               + ea.z * Ws[2 * H + j + 1] + ea.w * Ws[3 * H + j + 1], 0.0f);
    float m2 = fmaxf(hv.z + bs[j + 2] + ea.x * Ws[0 * H + j + 2] + ea.y * Ws[1 * H + j + 2]
                               + ea.z * Ws[2 * H + j + 2] + ea.w * Ws[3 * H + j + 2], 0.0f);
    float m3 = fmaxf(hv.w + bs[j + 3] + ea.x * Ws[0 * H + j + 3] + ea.y * Ws[1 * H + j + 3]
                               + ea.z * Ws[2 * H + j + 3] + ea.w * Ws[3 * H + j + 3], 0.0f);
    atomicAdd(arow + jj * 4 + 0, m0);
    atomicAdd(arow + jj * 4 + 1, m1);
    atomicAdd(arow + jj * 4 + 2, m2);
    atomicAdd(arow + jj * 4 + 3, m3);
  }
}

// ---------------------------------------------------------------------------
// Node MLP (per layer): h = relu(relu((h+agg) @ W1 + b1) @ W2 + b2), in place.
// fp32 WMMA; W1/W2 staged transposed in LDS; per-wave LDS bounce buffer to
// convert GEMM1's C-layout into GEMM2's A-layout. 4 waves/block, 64 rows.
// ---------------------------------------------------------------------------
__global__ __launch_bounds__(128) void gine_node_mlp(
    float* __restrict__ h, const float* __restrict__ agg,
    const float* __restrict__ W1, const float* __restrict__ b1,
    const float* __restrict__ W2, const float* __restrict__ b2) {
  __shared__ float W1t[H * H];  // [n][k]
  __shared__ float W2t[H * H];  // [n][k]
  __shared__ float b1s[H], b2s[H];
  __shared__ float tbuf[4][16 * H];  // per-wave 16x64 intermediate
  int tid = threadIdx.x;
  for (int i = tid; i < H * H; i += 128) {
    int k = i / H, n = i % H;
    W1t[n * H + k] = W1[i];
    W2t[n * H + k] = W2[i];
  }
  if (tid < H) { b1s[tid] = b1[tid]; b2s[tid] = b2[tid]; }
  __syncthreads();

  int wave = tid >> 5, lane = tid & 31;
  int m0 = blockIdx.x * 64 + wave * 16;
  if (m0 >= N_NODES) return;
  int mrow = lane & 15, half = lane >> 4;
  int row = m0 + mrow;
  float* tb = tbuf[wave];

  // GEMM1: t = relu((h+agg) @ W1 + b1)
  v8f acc[4] = {};
  #pragma unroll
  for (int kk = 0; kk < H / 4; ++kk) {
    int k = kk * 4 + half * 2;
    v2f ha = *(const v2f*)(h + row * H + k);
    v2f ga = *(const v2f*)(agg + row * H + k);
    v2f a = ha + ga;
    #pragma unroll
    for (int nt = 0; nt < 4; ++nt) {
      v2f b = *(const v2f*)(W1t + (nt * 16 + mrow) * H + k);
      acc[nt] = __builtin_amdgcn_wmma_f32_16x16x4_f32(
          false, a, false, b, (short)0, acc[nt], false, false);
    }
  }
  #pragma unroll
  for (int nt = 0; nt < 4; ++nt) {
    int n = nt * 16 + mrow;
    float bias = b1s[n];
    #pragma unroll
    for (int r = 0; r < 8; ++r)
      tb[(r + 8 * half) * H + n] = fmaxf(acc[nt][r] + bias, 0.0f);
  }
  asm volatile("s_wait_dscnt 0" ::: "memory");  // wave-local LDS RAW

  // GEMM2: h = relu(t @ W2 + b2)
  v8f acc2[4] = {};
  #pragma unroll
  for (int kk = 0; kk < H / 4; ++kk) {
    int k = kk * 4 + half * 2;
    v2f a = *(const v2f*)(tb + mrow * H + k);
    #pragma unroll
    for (int nt = 0; nt < 4; ++nt) {
      v2f b = *(const v2f*)(W2t + (nt * 16 + mrow) * H + k);
      acc2[nt] = __builtin_amdgcn_wmma_f32_16x16x4_f32(
          false, a, false, b, (short)0, acc2[nt], false, false);
    }
  }
  #pragma unroll
  for (int nt = 0; nt < 4; ++nt) {
    int n = nt * 16 + mrow;
    float bias = b2s[n];
    #pragma unroll
    for (int r = 0; r < 8; ++r)
      h[(m0 + r + 8 * half) * H + n] = fmaxf(acc2[nt][r] + bias, 0.0f);
  }
}

// ---------------------------------------------------------------------------
// Pooling: sums[batch[n]] += h[n]; cnts[batch[n]] += 1
// ---------------------------------------------------------------------------
__global__ __launch_bounds__(256) void gine_pool(
    const float* __restrict__ h, const int* __restrict__ batch,
    float* __restrict__ sums, float* __restrict__ cnts) {
  int idx = blockIdx.x * 256 + threadIdx.x;
  if (idx >= N_NODES * H) return;
  int n = idx >> 6, j = idx & 63;
  int b = batch[n];
  atomicAdd(&sums[b * H + j], h[idx]);
  if (j == 0) atomicAdd(&cnts[b], 1.0f);
}

// ---------------------------------------------------------------------------
// Final: mean pool -> layernorm -> out projection. One thread per graph.
// ---------------------------------------------------------------------------
__global__ __launch_bounds__(128) void gine_final(
    const float* __restrict__ sums, const float* __restrict__ cnts,
    const float* __restrict__ W_out, const float* __restrict__ b_out,
    float* __restrict__ out) {
  int g = blockIdx.x * 128 + threadIdx.x;
  if (g >= N_GRAPHS) return;
  float c = fmaxf(cnts[g], 1.0f);
  float gv[H];
  float mu = 0.0f;
  #pragma unroll
  for (int j = 0; j < H; ++j) { gv[j] = sums[g * H + j] / c; mu += gv[j]; }
  mu *= (1.0f / H);
  float var = 0.0f;
  #pragma unroll
  for (int j = 0; j < H; ++j) { float d = gv[j] - mu; var += d * d; }
  var *= (1.0f / H);
  float inv = rsqrtf(var + 1e-5f);
  float acc = 0.0f;
  #pragma unroll
  for (int j = 0; j < H; ++j) acc += (gv[j] - mu) * inv * W_out[j];
  out[g] = acc + b_out[0];
}

// ---------------------------------------------------------------------------
extern "C" void kernel_launch(void* const* d_in, const int* in_sizes, int n_in,
                              void* d_out, int out_size, void* d_ws, size_t ws_size,
                              hipStream_t stream) {
  const float* x         = (const float*)d_in[0];
  const int*   edge_index= (const int*)  d_in[1];
  const float* edge_attr = (const float*)d_in[2];
  const int*   batch     = (const int*)  d_in[3];
  const float* W_in      = (const float*)d_in[4];
  const float* b_in      = (const float*)d_in[5];
  const float* We        = (const float*)d_in[6];
  const float* be        = (const float*)d_in[7];
  const float* W1        = (const float*)d_in[8];
  const float* b1        = (const float*)d_in[9];
  const float* W2        = (const float*)d_in[10];
  const float* b2        = (const float*)d_in[11];
  const float* W_out     = (const float*)d_in[12];
  const float* b_out     = (const float*)d_in[13];
  float* out = (float*)d_out;

  // Workspace layout (floats): h[N*H] | agg[N*H] | sums[G*H] | cnts[G]
  float* h    = (float*)d_ws;
  float* agg  = h   + (size_t)N_NODES * H;
  float* sums = agg + (size_t)N_NODES * H;
  float* cnts = sums + (size_t)N_GRAPHS * H;

  const int* src = edge_index;
  const int* dst = edge_index + N_EDGES;

  gine_in_proj<<<(N_NODES + 63) / 64, 128, 0, stream>>>(x, W_in, b_in, h);

  for (int l = 0; l < N_LAYERS; ++l) {
    hipMemsetAsync(agg, 0, (size_t)N_NODES * H * sizeof(float), stream);
    gine_edge<<<(N_EDGES + 63) / 64, 256, 0, stream>>>(
        h, src, dst, edge_attr, We + l * EDGE_DIM * H, be + l * H, agg);
    gine_node_mlp<<<(N_NODES + 63) / 64, 128, 0, stream>>>(
        h, agg, W1 + l * H * H, b1 + l * H, W2 + l * H * H, b2 + l * H);
  }

  hipMemsetAsync(sums, 0, (size_t)(N_GRAPHS * H + N_GRAPHS) * sizeof(float), stream);
  gine_pool<<<(N_NODES * H + 255) / 256, 256, 0, stream>>>(h, batch, sums, cnts);
  gine_final<<<(N_GRAPHS + 127) / 128, 128, 0, stream>>>(sums, cnts, W_out, b_out, out);
}